// VectorQuantize_56848187130000
// MI455X (gfx1250) — compile-verified
//
#include <hip/hip_runtime.h>
#include <hip/hip_bf16.h>
#include <math.h>

// ---- problem constants (from reference setup_inputs) ----
#define B_  16
#define D_  1024
#define T_  4096
#define CD_ 8
#define CS_ 1024
#define TT_ 256          // timesteps per block (1 per thread)

// d_out layout (floats, concatenated reference outputs)
#define OUT_N   (B_ * D_ * T_)          // 67108864
#define CL_OFF  OUT_N                    // commitment loss [16]
#define QL_OFF  (CL_OFF + 16)            // codebook loss   [16]
#define IDX_OFF (QL_OFF + 16)            // indices [16*4096] as float
#define ZE_OFF  (IDX_OFF + B_ * T_)      // z_e [16*8*4096]

typedef __attribute__((ext_vector_type(2))) float v2f;
typedef __attribute__((ext_vector_type(8))) float v8f;

// ============================================================
// Prep: weight-norm w_in (-> [d][8] transposed), weight-norm w_out [d][8],
// l2-normalized codebook [cs][8]; zero the 32 loss accumulators.
// NOTE: the three tables are written CONTIGUOUSLY in ws (24576 floats)
// so the main kernel can async-copy them to LDS in one sweep.
// ============================================================
__global__ __launch_bounds__(256)
void vq_prep(const float* __restrict__ inV, const float* __restrict__ inG,
             const float* __restrict__ outV, const float* __restrict__ outG,
             const float* __restrict__ cb,
             float* __restrict__ w_in_t, float* __restrict__ cb_n,
             float* __restrict__ w_out, float* __restrict__ loss_zero)
{
    __shared__ float s_rin[CD_];
    const int tid  = threadIdx.x;
    const int lane = tid & 31;
    const int wv   = tid >> 5;           // 8 waves -> one per in-proj row

    // per-row ||v|| for in_proj (rows of length 1024), one wave each
    {
        float p = 0.f;
        for (int d = lane; d < D_; d += 32) {
            float v = inV[wv * D_ + d];
            p = fmaf(v, v, p);
        }
        #pragma unroll
        for (int st = 16; st; st >>= 1) p += __shfl_xor(p, st);
        if (lane == 0) s_rin[wv] = inG[wv] / sqrtf(p);
    }
    if (tid < 32) loss_zero[tid] = 0.f;   // commitment[16] + codebook[16]
    __syncthreads();

    // w_in transposed to [d][8] for LDS-broadcast-friendly access
    for (int i = tid; i < D_ * CD_; i += 256) {
        int d = i >> 3, c = i & 7;
        w_in_t[i] = inV[c * D_ + d] * s_rin[c];
    }
    // normalized codebook + weight-normed out projection (row norms over 8)
    for (int r = tid; r < CS_; r += 256) {
        float v[CD_], ss = 0.f;
        #pragma unroll
        for (int c = 0; c < CD_; ++c) { v[c] = cb[r * CD_ + c]; ss = fmaf(v[c], v[c], ss); }
        float s = 1.f / fmaxf(sqrtf(ss), 1e-12f);
        #pragma unroll
        for (int c = 0; c < CD_; ++c) cb_n[r * CD_ + c] = v[c] * s;

        float u[CD_]; ss = 0.f;
        #pragma unroll
        for (int c = 0; c < CD_; ++c) { u[c] = outV[r * CD_ + c]; ss = fmaf(u[c], u[c], ss); }
        float s2 = outG[r] / sqrtf(ss);
        #pragma unroll
        for (int c = 0; c < CD_; ++c) w_out[r * CD_ + c] = u[c] * s2;
    }
}

// Async global->LDS copy of 16B per lane (bypasses VGPRs, tracked by ASYNCcnt).
__device__ __forceinline__ void async_b128_to_lds(const float* g, const float* l) {
    unsigned loff = (unsigned)(uintptr_t)l;      // low 32 bits = LDS byte offset
    asm volatile("global_load_async_to_lds_b128 %0, %1, off"
                 :: "v"(loff), "v"(g) : "memory");
}

// ============================================================
// Fused VQ: in-proj -> normalize -> WMMA NN search -> losses -> out-proj
// grid = B*(T/TT) = 256 blocks, 256 threads (8 waves), 1 timestep/thread.
// ============================================================
__global__ __launch_bounds__(256)
void vq_fused(const float* __restrict__ z, const float* __restrict__ in_b,
              const float* __restrict__ codebook,
              const float* __restrict__ tables,   // [w_in_t | cb_n | w_out] 24576 f
              const float* __restrict__ out_b,
              float* __restrict__ out, float* __restrict__ loss_c,
              float* __restrict__ loss_q, float* __restrict__ idx_out,
              float* __restrict__ ze_out)
{
    extern __shared__ float smem[];
    float* s_win  = smem;                  // [1024][8]  32 KB
    float* s_cb   = smem + 8192;           // [1024][8]  32 KB (normalized)
    float* s_wout = smem + 16384;          // [1024][8]  32 KB
    float* s_bias = smem + 24576;          // [1024]      4 KB
    __shared__ float s_loss;

    const int tid = threadIdx.x;
    const int b   = blockIdx.x >> 4;
    const int t   = (blockIdx.x & 15) * TT_ + tid;

    // ---- async LDS fill: 96 KB of tables + 4 KB bias, B128 per lane ----
    #pragma unroll 4
    for (int i = tid * 4; i < 3 * D_ * CD_; i += 4 * 256) {
        async_b128_to_lds(tables + i, smem + i);
    }
    async_b128_to_lds(out_b + tid * 4, s_bias + tid * 4);
    if (tid == 0) s_loss = 0.f;
    asm volatile("s_wait_asynccnt 0" ::: "memory");
    __syncthreads();

    // ---- in-projection: z_e[c] = b[c] + sum_d w_in[c][d]*z[b][d][t] ----
    float acc[CD_];
    #pragma unroll
    for (int c = 0; c < CD_; ++c) acc[c] = in_b[c];
    const float* zb = z + (size_t)b * D_ * T_ + t;
    for (int d = 0; d < D_; ++d) {
        float zv = zb[(size_t)d * T_];
        int dp = d + 8 < D_ ? d + 8 : d;
        __builtin_prefetch(zb + (size_t)dp * T_, 0, 0);   // global_prefetch_b8
        #pragma unroll
        for (int c = 0; c < CD_; ++c) acc[c] = fmaf(s_win[d * CD_ + c], zv, acc[c]);
    }
    // emit z_e output
    {
        float* zeb = ze_out + (size_t)b * CD_ * T_ + t;
        #pragma unroll
        for (int c = 0; c < CD_; ++c) zeb[(size_t)c * T_] = acc[c];
    }

    // ---- l2 normalize encoding ----
    float ss = 0.f;
    #pragma unroll
    for (int c = 0; c < CD_; ++c) ss = fmaf(acc[c], acc[c], ss);
    const float inv = 1.f / fmaxf(sqrtf(ss), 1e-12f);
    float en[CD_];
    #pragma unroll
    for (int c = 0; c < CD_; ++c) en[c] = acc[c] * inv;

    // ---- WMMA cosine-sim NN search: sim = E_n(16t x 8) * C_n^T(8 x 16s) ----
    const int lane = tid & 31;
    const int l16  = lane & 15;
    const int hi   = lane >> 4;          // lane half selects K pair per ISA A-layout
    int myIdx = 0;

    #pragma unroll
    for (int h = 0; h < 2; ++h) {        // two 16-timestep halves per wave
        const int srcLane = h * 16 + l16;
        // Build A (16x4 f32): lanes<16 carry K={k0,k0+1}, lanes>=16 K={k0+2,k0+3}
        v2f a0, a1;
        {
            float e0 = __shfl(en[0], srcLane), e1 = __shfl(en[1], srcLane);
            float e2 = __shfl(en[2], srcLane), e3 = __shfl(en[3], srcLane);
            float e4 = __shfl(en[4], srcLane), e5 = __shfl(en[5], srcLane);
            float e6 = __shfl(en[6], srcLane), e7 = __shfl(en[7], srcLane);
            a0.x = hi ? e2 : e0;  a0.y = hi ? e3 : e1;
            a1.x = hi ? e6 : e4;  a1.y = hi ? e7 : e5;
        }
        float bestV[8]; int bestI[8];
        #pragma unroll
        for (int r = 0; r < 8; ++r) { bestV[r] = -3.402823466e38f; bestI[r] = 0; }

        for (int s0 = 0; s0 < CS_; s0 += 16) {
            // B (4x16 f32): lane L carries code row s0+l16, K pair selected by hi
            const float* cbr = s_cb + (s0 + l16) * CD_ + hi * 2;
            v2f b0; b0.x = cbr[0]; b0.y = cbr[1];
            v2f b1; b1.x = cbr[4]; b1.y = cbr[5];
            v8f sim = {};
            sim = __builtin_amdgcn_wmma_f32_16x16x4_f32(false, a0, false, b0,
                                                        (short)0, sim, false, false);
            sim = __builtin_amdgcn_wmma_f32_16x16x4_f32(false, a1, false, b1,
                                                        (short)0, sim, false, false);
            const int sidx = s0 + l16;
            #pragma unroll
            for (int r = 0; r < 8; ++r) {       // branchless running argmax
                float v = sim[r];
                bool better = v > bestV[r];      // strict: keeps first max (s ascending)
                bestV[r] = better ? v : bestV[r];
                bestI[r] = better ? sidx : bestI[r];
            }
        }
        // reduce (val,idx) across the 16 lanes of each half; tie -> smaller idx
        #pragma unroll
        for (int r = 0; r < 8; ++r) {
            #pragma unroll
            for (int st = 1; st < 16; st <<= 1) {
                float ov = __shfl_xor(bestV[r], st);
                int   oi = __shfl_xor(bestI[r], st);
                bool take = (ov > bestV[r]) || (ov == bestV[r] && oi < bestI[r]);
                bestV[r] = take ? ov : bestV[r];
                bestI[r] = take ? oi : bestI[r];
            }
            // C layout: lanes0-15 VGPR r = M=r ; lanes16-31 VGPR r = M=8+r
            int iLo = __shfl(bestI[r], 0);
            int iHi = __shfl(bestI[r], 16);
            if ((lane >> 4) == h) {
                if (l16 == r)     myIdx = iLo;
                if (l16 == r + 8) myIdx = iHi;
            }
        }
    }

    // ---- gather code, losses, straight-through ----
    const float* cbRow = codebook + (size_t)myIdx * CD_;
    float st8[CD_];
    float diff2 = 0.f;
    #pragma unroll
    for (int c = 0; c < CD_; ++c) {
        float q = cbRow[c];
        float dlt = acc[c] - q;
        diff2 = fmaf(dlt, dlt, diff2);
        st8[c] = acc[c] + (q - acc[c]);   // match reference z_e + sg(z_q - z_e) rounding
    }
    idx_out[b * T_ + t] = (float)myIdx;

    atomicAdd(&s_loss, diff2);
    __syncthreads();
    if (tid == 0) {
        float contrib = s_loss * (1.0f / (float)(CD_ * T_));
        atomicAdd(&loss_c[b], contrib);   // commitment
        atomicAdd(&loss_q[b], contrib);   // codebook (numerically identical)
    }

    // ---- out-projection: out[d][t] = bias[d] + sum_c w_out[d][c]*st[c] ----
    float* ob = out + (size_t)b * D_ * T_ + t;
    for (int d = 0; d < D_; ++d) {
        const float* wr = s_wout + d * CD_;
        float val = s_bias[d];
        #pragma unroll
        for (int c = 0; c < CD_; ++c) val = fmaf(wr[c], st8[c], val);
        __builtin_nontemporal_store(val, ob + (size_t)d * T_);
    }
}

extern "C" void kernel_launch(void* const* d_in, const int* in_sizes, int n_in,
                              void* d_out, int out_size, void* d_ws, size_t ws_size,
                              hipStream_t stream) {
    const float* z    = (const float*)d_in[0];
    const float* inV  = (const float*)d_in[1];
    const float* inG  = (const float*)d_in[2];
    const float* inB  = (const float*)d_in[3];
    const float* outV = (const float*)d_in[4];
    const float* outG = (const float*)d_in[5];
    const float* outB = (const float*)d_in[6];
    const float* cb   = (const float*)d_in[7];

    float* out    = (float*)d_out;
    float* w_in_t = (float*)d_ws;          // 8192 f
    float* cb_n   = w_in_t + 8192;         // 8192 f
    float* w_out  = w_in_t + 16384;        // 8192 f  (contiguous 24576-f table block)

    float* o_loss = out + CL_OFF;          // 32 floats (commitment + codebook)
    float* o_idx  = out + IDX_OFF;
    float* o_ze   = out + ZE_OFF;

    vq_prep<<<dim3(1), dim3(256), 0, stream>>>(inV, inG, outV, outG, cb,
                                               w_in_t, cb_n, w_out, o_loss);

    const size_t lds = (size_t)(3 * D_ * CD_ + D_) * sizeof(float);  // 100 KB
    vq_fused<<<dim3(B_ * (T_ / TT_)), dim3(TT_), lds, stream>>>(
        z, inB, cb, w_in_t, outB,
        out, o_loss, o_loss + 16, o_idx, o_ze);
}